// Attribute_Decoder_54107997995613
// MI455X (gfx1250) — compile-verified
//
#include <hip/hip_runtime.h>

typedef __attribute__((ext_vector_type(2))) float v2f;
typedef __attribute__((ext_vector_type(8))) float v8f;

#define NHID 128

// C[M x ncols] = act(A)[M x K] @ W[K x ncols], K multiple of 4, ncols multiple of 128.
// One wave computes a 16-row x 128-col strip with V_WMMA_F32_16X16X4_F32.
// RELU: apply relu to A elements on load (used for layer 2, where A = agg1 + b1).
template <bool RELU>
__global__ __launch_bounds__(128) void gcn_gemm_wmma(
    const float* __restrict__ A, const float* __restrict__ W,
    float* __restrict__ C, int M, int K, int ncols)
{
    const int lane  = threadIdx.x & 31;
    const int wave  = threadIdx.x >> 5;
    const int mtile = blockIdx.x * 4 + wave;
    if (mtile * 16 >= M) return;          // wave-uniform: EXEC stays all-1s
    const int m0 = mtile * 16;
    const int n0 = blockIdx.y * 128;
    const int hi = lane >> 4;             // 0: K=k,k+1   1: K=k+2,k+3
    const int lm = lane & 15;             // row (A) / col (B,C) within tile

    v8f acc[8] = {};
    const float* arow = A + (size_t)(m0 + lm) * K;

    for (int k = 0; k < K; k += 4) {
        // A fragment 16x4 fp32: lanes 0-15 hold K=k..k+1, lanes 16-31 K=k+2..k+3
        v2f a = *(const v2f*)(arow + k + 2 * hi);
        if (RELU) { a.x = fmaxf(a.x, 0.f); a.y = fmaxf(a.y, 0.f); }
        const float* w0 = W + (size_t)(k + 2 * hi) * ncols + n0 + lm;
        const float* w1 = w0 + ncols;
#pragma unroll
        for (int t = 0; t < 8; ++t) {
            v2f b;
            b.x = w0[t * 16];             // B row k+2*hi,   col n0+t*16+lm
            b.y = w1[t * 16];             // B row k+2*hi+1, col n0+t*16+lm
            acc[t] = __builtin_amdgcn_wmma_f32_16x16x4_f32(
                false, a, false, b, (short)0, acc[t], false, false);
        }
    }

    // C/D layout: VGPR i -> row m0+i (lanes 0-15) / m0+i+8 (lanes 16-31), col = lm
#pragma unroll
    for (int t = 0; t < 8; ++t) {
#pragma unroll
        for (int i = 0; i < 8; ++i) {
            C[(size_t)(m0 + i + 8 * hi) * ncols + (n0 + t * 16 + lm)] = acc[t][i];
        }
    }
}

// One wave per edge: agg[dst[e]] += support[src[e]] * w[e].
// Edge metadata is wave-uniform (scalarizes); gather is float4 (b128) per lane;
// scatter uses hardware global_atomic_add_f32 into the L2-resident aggregator.
__global__ __launch_bounds__(256) void gcn_scatter(
    const float* __restrict__ sup,
    const long long* __restrict__ src,
    const long long* __restrict__ dst,
    const float* __restrict__ w,
    float* __restrict__ agg, int E, int D)
{
    const int lane = threadIdx.x & 31;
    const int e    = blockIdx.x * (blockDim.x >> 5) + (threadIdx.x >> 5);
    if (e >= E) return;
    const long long s  = src[e];
    const long long t  = dst[e];
    const float     wt = w[e];
    const float4* srow = (const float4*)(sup + (size_t)s * D);
    float*        drow = agg + (size_t)t * D;
    const int nchunk = D >> 7;            // float4 chunks per lane (1 for D=128, 2 for 256)
#pragma unroll
    for (int c = 0; c < 2; ++c) {
        if (c >= nchunk) break;
        const int idx = c * 32 + lane;
        float4 v = srow[idx];
        float* p = drow + idx * 4;
        unsafeAtomicAdd(p + 0, v.x * wt);
        unsafeAtomicAdd(p + 1, v.y * wt);
        unsafeAtomicAdd(p + 2, v.z * wt);
        unsafeAtomicAdd(p + 3, v.w * wt);
    }
}

// y[i] = b[col(i)] broadcast (float4-vectorized); replaces the memset so the
// scatter accumulates directly on top of the bias.
__global__ void gcn_fill_bias(float4* __restrict__ y, const float4* __restrict__ b,
                              size_t total4, int colmask4)
{
    const size_t stride = (size_t)gridDim.x * blockDim.x;
    for (size_t i = (size_t)blockIdx.x * blockDim.x + threadIdx.x; i < total4; i += stride)
        y[i] = b[i & (size_t)colmask4];
}

// In-place ReLU, float4-vectorized (final pass on d_out).
__global__ void gcn_relu4(float4* __restrict__ y, size_t total4)
{
    const size_t stride = (size_t)gridDim.x * blockDim.x;
    for (size_t i = (size_t)blockIdx.x * blockDim.x + threadIdx.x; i < total4; i += stride) {
        float4 v = y[i];
        v.x = fmaxf(v.x, 0.f); v.y = fmaxf(v.y, 0.f);
        v.z = fmaxf(v.z, 0.f); v.w = fmaxf(v.w, 0.f);
        y[i] = v;
    }
}

extern "C" void kernel_launch(void* const* d_in, const int* in_sizes, int n_in,
                              void* d_out, int out_size, void* d_ws, size_t ws_size,
                              hipStream_t stream)
{
    const float*     x   = (const float*)d_in[0];
    const float*     ew  = (const float*)d_in[1];
    const float*     W1  = (const float*)d_in[2];
    const float*     b1  = (const float*)d_in[3];
    const float*     W2  = (const float*)d_in[4];
    const float*     b2  = (const float*)d_in[5];
    const long long* ei  = (const long long*)d_in[6];

    const int N  = in_sizes[0] / NHID;     // 100000
    const int E  = in_sizes[1];            // 1600000
    const int NF = out_size / N;           // 256

    const long long* src  = ei;            // edge_index[0]
    const long long* dstv = ei + E;        // edge_index[1]

    // Workspace: [agg1: N*NHID] [support1/support2 (overlapped): N*NF]
    float* A1 = (float*)d_ws;
    float* S1 = A1 + (size_t)N * NHID;
    float* S2 = S1;                        // S1 dead before S2 is written

    const int mtiles = (N + 15) / 16;
    const int gx     = (mtiles + 3) / 4;
    const int eblk   = (E + 7) / 8;        // 8 edges (waves) per 256-thread block

    // ---- layer 1: A1 = b1 + scatter(x @ W1)   (relu deferred into GEMM2) ----
    gcn_fill_bias<<<2048, 256, 0, stream>>>((float4*)A1, (const float4*)b1,
                                            (size_t)N * NHID / 4, NHID / 4 - 1);
    gcn_gemm_wmma<false><<<dim3(gx, NHID / 128), 128, 0, stream>>>(x, W1, S1, N, NHID, NHID);
    gcn_scatter<<<eblk, 256, 0, stream>>>(S1, src, dstv, ew, A1, E, NHID);

    // ---- layer 2: out = relu( b2 + scatter( relu(A1) @ W2 ) ) ----
    gcn_fill_bias<<<2048, 256, 0, stream>>>((float4*)d_out, (const float4*)b2,
                                            (size_t)N * NF / 4, NF / 4 - 1);
    gcn_gemm_wmma<true><<<dim3(gx, NF / 128), 128, 0, stream>>>(A1, W2, S2, N, NHID, NF);
    gcn_scatter<<<eblk, 256, 0, stream>>>(S2, src, dstv, ew, (float*)d_out, E, NF);
    gcn_relu4<<<2048, 256, 0, stream>>>((float4*)d_out, (size_t)N * NF / 4);
}